// SFS_18064632447437
// MI455X (gfx1250) — compile-verified
//
#include <hip/hip_runtime.h>
#include <math.h>

#define S_FRAMES 64
#define CCH      384
#define HW       3136      // 56*56
#define STOP     16
#define F2       32        // 2*S_TOP
#define CNUM     48        // C / C_DIV
#define BN_EPS   1e-5f

typedef float v2f __attribute__((ext_vector_type(2)));
typedef float v8f __attribute__((ext_vector_type(8)));

// ---------------------------------------------------------------------------
// Kernel 1: global average pool per (s, c) plane.  One block per plane,
// float4 loads, LDS tree reduction.  Pure-bandwidth pass (308 MB read).
// ---------------------------------------------------------------------------
__global__ void SFS_pool_kernel(const float* __restrict__ x, float* __restrict__ xg) {
    const int plane = blockIdx.x;                     // 0 .. S*C-1
    const float4* p = (const float4*)(x + (size_t)plane * HW);
    const int tid = threadIdx.x;
    float sum = 0.f;
    for (int j = tid; j < HW / 4; j += 256) {         // 784 float4 per plane
        float4 v = p[j];
        sum += v.x + v.y + v.z + v.w;
    }
    __shared__ float red[256];
    red[tid] = sum;
    __syncthreads();
    for (int off = 128; off > 0; off >>= 1) {
        if (tid < off) red[tid] += red[tid + off];
        __syncthreads();
    }
    if (tid == 0) xg[plane] = red[0] * (1.0f / (float)HW);
}

// ---------------------------------------------------------------------------
// Kernel 2: per-frame score = xg @ w_lin, then stable top-16 by score,
// emitted in ascending (temporal) order.  Negligible work: single block.
// ---------------------------------------------------------------------------
__global__ void SFS_topk_kernel(const float* __restrict__ xg,
                                const float* __restrict__ w_lin,
                                int* __restrict__ idx_top) {
    __shared__ float sc[S_FRAMES];
    const int tid = threadIdx.x;
    if (tid < S_FRAMES) {
        float d = 0.f;
        for (int c = 0; c < CCH; ++c) d += xg[tid * CCH + c] * w_lin[c];
        sc[tid] = d;
    }
    __syncthreads();
    if (tid == 0) {
        // stable argsort(-score) semantics: ties broken by lower index
        int n = 0;
        for (int s = 0; s < S_FRAMES && n < STOP; ++s) {
            int rank = 0;
            const float v = sc[s];
            for (int j = 0; j < S_FRAMES; ++j)
                if (sc[j] > v || (sc[j] == v && j < s)) rank++;
            if (rank < STOP) idx_top[n++] = s;
        }
    }
}

// ---------------------------------------------------------------------------
// Kernel 3: kernel-generating MLP via V_WMMA_F32_16X16X4_F32 (exact fp32).
// One wave (32 threads) per 16-channel tile.  h1(16x32) = sls(16x16) @ w1^T,
// accumulated as 2 N-tiles x 4 K-chunks of 16x16x4 WMMA.  Then BN(eval) +
// ReLU per lane, stage through LDS, tiny 32->3 GEMM + softmax per channel.
//
// f32 WMMA layouts (ISA 7.12.2):
//   A 16x4 : lane L: M=L%16; v[0]=K(0|2), v[1]=K(1|3)  (|: lanes 16-31)
//   B 4x16 : lane L: N=L%16; v[0]=K(0|2), v[1]=K(1|3)
//   C/D    : VGPR r, lane L: M=r+8*(L>=16), N=L%16
// ---------------------------------------------------------------------------
__global__ void SFS_mlp_wmma_kernel(const float* __restrict__ xg,
                                    const int* __restrict__ idx_top,
                                    const float* __restrict__ w1,       // (32,16)
                                    const float* __restrict__ bn_gamma, // (32,)
                                    const float* __restrict__ bn_beta,
                                    const float* __restrict__ bn_mean,
                                    const float* __restrict__ bn_var,
                                    const float* __restrict__ w2,       // (3,32)
                                    float* __restrict__ convk) {        // (384,3)
    const int lane = threadIdx.x;         // 0..31, full wave, EXEC all-ones
    const int m    = lane & 15;
    const int hi   = lane >> 4;
    const int c    = blockIdx.x * 16 + m; // this lane's A-matrix row (channel)

    __shared__ int topi[STOP];
    if (lane < STOP) topi[lane] = idx_top[lane];
    __syncthreads();

    v8f acc0 = {};   // N-tile 0: features 0..15
    v8f acc1 = {};   // N-tile 1: features 16..31
    for (int k = 0; k < 4; ++k) {
        const int t0 = 4 * k + 2 * hi;     // K index held in v[0] for this lane
        v2f a, b0, b1;
        // A(M=c_local, K=t) = sls[c][t] = xg[idx_top[t]][c]
        a[0] = xg[(size_t)topi[t0]     * CCH + c];
        a[1] = xg[(size_t)topi[t0 + 1] * CCH + c];
        // B(K=t, N=n) = w1^T[t][n] = w1[n][t]
        b0[0] = w1[m * STOP + t0];
        b0[1] = w1[m * STOP + t0 + 1];
        b1[0] = w1[(16 + m) * STOP + t0];
        b1[1] = w1[(16 + m) * STOP + t0 + 1];
        acc0 = __builtin_amdgcn_wmma_f32_16x16x4_f32(false, a, false, b0,
                                                     (short)0, acc0, false, false);
        acc1 = __builtin_amdgcn_wmma_f32_16x16x4_f32(false, a, false, b1,
                                                     (short)0, acc1, false, false);
    }

    // BN (eval) + ReLU.  Feature index is lane-constant per N-tile.
    const float sc0 = bn_gamma[m]      * rsqrtf(bn_var[m]      + BN_EPS);
    const float sc1 = bn_gamma[16 + m] * rsqrtf(bn_var[16 + m] + BN_EPS);
    const float mu0 = bn_mean[m],     mu1 = bn_mean[16 + m];
    const float be0 = bn_beta[m],     be1 = bn_beta[16 + m];

    __shared__ float h1s[16 * F2];        // 16 channels x 32 features
    #pragma unroll
    for (int r = 0; r < 8; ++r) {
        const int row = r + 8 * hi;       // channel row within tile
        float h0 = (acc0[r] - mu0) * sc0 + be0;
        float h1 = (acc1[r] - mu1) * sc1 + be1;
        h1s[row * F2 + m]      = h0 > 0.f ? h0 : 0.f;
        h1s[row * F2 + 16 + m] = h1 > 0.f ? h1 : 0.f;
    }
    __syncthreads();

    // Second linear (32 -> 3) + softmax, one channel per lane.
    if (lane < 16) {
        const int ch = blockIdx.x * 16 + lane;
        float o[3];
        #pragma unroll
        for (int j = 0; j < 3; ++j) {
            float d = 0.f;
            for (int f = 0; f < F2; ++f) d += h1s[lane * F2 + f] * w2[j * F2 + f];
            o[j] = d;
        }
        const float mx = fmaxf(o[0], fmaxf(o[1], o[2]));
        const float e0 = expf(o[0] - mx), e1 = expf(o[1] - mx), e2 = expf(o[2] - mx);
        const float inv = 1.f / (e0 + e1 + e2);
        convk[ch * 3 + 0] = e0 * inv;
        convk[ch * 3 + 1] = e1 * inv;
        convk[ch * 3 + 2] = e2 * inv;
    }
}

// ---------------------------------------------------------------------------
// Kernel 4: fused channel-shift + depthwise temporal conv (k=3, pad=1).
// One block per (channel, 256-pixel tile); rolling register window over s so
// every input element is read from HBM exactly once (no 3x reuse penalty).
// xl[t] = x[t+shift] with zeroing both at the shift boundary and conv pad.
// out[s] = k0*xl[s-1] + k1*xl[s] + k2*xl[s+1].
// ---------------------------------------------------------------------------
__global__ void SFS_shiftconv_kernel(const float* __restrict__ x,
                                     const float* __restrict__ convk,
                                     float* __restrict__ out) {
    const int c = blockIdx.y;
    const int p = blockIdx.x * 256 + threadIdx.x;
    const bool active = (p < HW);
    const int shift = (c < CNUM) ? 1 : ((c < 2 * CNUM) ? -1 : 0);
    const float k0 = convk[c * 3 + 0];
    const float k1 = convk[c * 3 + 1];
    const float k2 = convk[c * 3 + 2];
    const size_t fs = (size_t)CCH * HW;            // frame stride (elements)
    const float* xc = x   + (size_t)c * HW + p;
    float*       oc = out + (size_t)c * HW + p;

    auto loadXL = [&](int t) -> float {
        const int src = t + shift;
        if (!active || (unsigned)t >= S_FRAMES || (unsigned)src >= S_FRAMES)
            return 0.f;
        return xc[(size_t)src * fs];
    };

    float a = 0.f;              // xl[s-1]  (conv zero-pad at s=-1)
    float b = loadXL(0);        // xl[s]
    for (int s = 0; s < S_FRAMES; ++s) {
        const float d = loadXL(s + 1);
        if (active) oc[(size_t)s * fs] = k0 * a + k1 * b + k2 * d;
        a = b; b = d;
    }
}

// ---------------------------------------------------------------------------
extern "C" void kernel_launch(void* const* d_in, const int* in_sizes, int n_in,
                              void* d_out, int out_size, void* d_ws, size_t ws_size,
                              hipStream_t stream) {
    const float* x        = (const float*)d_in[0];
    const float* w_lin    = (const float*)d_in[1];
    const float* w1       = (const float*)d_in[2];
    const float* bn_gamma = (const float*)d_in[3];
    const float* bn_beta  = (const float*)d_in[4];
    const float* bn_mean  = (const float*)d_in[5];
    const float* bn_var   = (const float*)d_in[6];
    const float* w2       = (const float*)d_in[7];
    float* out = (float*)d_out;

    // Workspace layout (all fully rewritten every call; no cross-call state):
    float* xg    = (float*)d_ws;                 // 64*384 floats
    float* convk = xg + S_FRAMES * CCH;          // 384*3 floats
    int*   idxt  = (int*)(convk + CCH * 3);      // 16 ints

    SFS_pool_kernel<<<S_FRAMES * CCH, 256, 0, stream>>>(x, xg);
    SFS_topk_kernel<<<1, 64, 0, stream>>>(xg, w_lin, idxt);
    SFS_mlp_wmma_kernel<<<CCH / 16, 32, 0, stream>>>(xg, idxt, w1, bn_gamma,
                                                     bn_beta, bn_mean, bn_var,
                                                     w2, convk);
    dim3 grid((HW + 255) / 256, CCH);
    SFS_shiftconv_kernel<<<grid, 256, 0, stream>>>(x, convk, out);
}